// R_transformations_18133351924158
// MI455X (gfx1250) — compile-verified
//
#include <hip/hip_runtime.h>
#include <math.h>

// ---------------------------------------------------------------------------
// Complex FNO U-Net forward for MI455X (gfx1250, wave32).
//   * batch-of-GEMMs M=16(batch) x K=ci x N=co per spatial location via
//     V_WMMA_F32_16X16X4_F32 (4 accumulator chains for the complex product)
//   * weight-bandwidth bound (~410MB f32): local-layer GEMM fuses the
//     (ic,oc,x,y) -> (K x N per-location) transpose through LDS, streaming
//     weights ONCE with GLOBAL_LOAD_ASYNC_TO_LDS (ASYNCcnt) when available.
// ---------------------------------------------------------------------------

typedef __attribute__((ext_vector_type(2))) float v2f;
typedef __attribute__((ext_vector_type(8))) float v8f;

#if defined(__has_builtin)
#if __has_builtin(__builtin_amdgcn_global_load_async_to_lds_b32)
#define HAVE_ASYNC_LDS 1
#endif
#endif
#ifndef HAVE_ASYNC_LDS
#define HAVE_ASYNC_LDS 0
#endif

typedef __attribute__((address_space(1))) int as1_int;
typedef __attribute__((address_space(3))) int as3_int;

__device__ __forceinline__ float gelu_exact(float x) {
    return 0.5f * x * (1.0f + erff(x * 0.70710678118654752f));
}

#define WMMA_F32(a, b, c) \
    __builtin_amdgcn_wmma_f32_16x16x4_f32(false, (a), false, (b), (short)0, (c), false, false)

// ---------------------------------------------------------------------------
// l0 GEMM (shared 76->64 weights, already coalesced in native (ic,oc) layout).
// One block per location, one wave per 16-wide N tile.
// WMMA f32 16x16x4 layouts (ISA 7.12.2):
//   A: vgpr0 = A[M=l][K=k0+2*half], vgpr1 = K+1   (half = lane>=16)
//   B: vgpr0 = B[K=k0+2*half][N=l], vgpr1 = K+1
//   D: lane<16 -> (M=v, N=l); lane>=16 -> (M=v+8, N=l)
// ---------------------------------------------------------------------------
__global__ void gemm_cplx_shared(const float* __restrict__ Ar, const float* __restrict__ Ai,
                                 const float* __restrict__ Wr, const float* __restrict__ Wi,
                                 const float* __restrict__ Br, const float* __restrict__ Bi,
                                 float* __restrict__ Yr, float* __restrict__ Yi,
                                 int ci, int co)
{
    const int loc  = blockIdx.x;
    const int lane = threadIdx.x;
    const int half = lane >> 4;
    const int l    = lane & 15;
    const int n0   = threadIdx.y << 4;

    const float* __restrict__ Abr = Ar + (size_t)loc * ci * 16;
    const float* __restrict__ Abi = Ai + (size_t)loc * ci * 16;

    v8f acc_rr = {}, acc_ii = {}, acc_ri = {}, acc_ir = {};

    for (int k0 = 0; k0 < ci; k0 += 4) {
        const int ka = k0 + (half << 1);
        v2f a_r = { Abr[ka * 16 + l], Abr[(ka + 1) * 16 + l] };
        v2f a_i = { Abi[ka * 16 + l], Abi[(ka + 1) * 16 + l] };
        const size_t w0 = (size_t)ka * co + n0 + l;
        v2f b_r = { Wr[w0], Wr[w0 + co] };
        v2f b_i = { Wi[w0], Wi[w0 + co] };
        acc_rr = WMMA_F32(a_r, b_r, acc_rr);
        acc_ii = WMMA_F32(a_i, b_i, acc_ii);
        acc_ri = WMMA_F32(a_r, b_i, acc_ri);
        acc_ir = WMMA_F32(a_i, b_r, acc_ir);
    }

    const int oc = n0 + l;
    const float bre = Br[oc], bim = Bi[oc];
    float* __restrict__ Yrp = Yr + ((size_t)loc * co + n0) * 16;
    float* __restrict__ Yip = Yi + ((size_t)loc * co + n0) * 16;
#pragma unroll
    for (int v = 0; v < 8; ++v) {
        const int m = v + (half << 3);
        Yrp[l * 16 + m] = acc_rr[v] - acc_ii[v] + bre;
        Yip[l * 16 + m] = acc_ri[v] + acc_ir[v] + bim;
    }
}

// ---------------------------------------------------------------------------
// Fused local-layer GEMM: per-location weights streamed ONCE from native
// (ic,oc,x,y) layout. Block = (32,16): 16 waves = 16 consecutive y locations
// of row x. Per K-chunk of 8, all 512 threads cooperatively stage
// 2(planes)*8(k)*16(n)*16(y) dwords into LDS (global reads = contiguous
// 16-y runs) via async-to-LDS loads, then each wave reads its transposed
// 4x16 B-tiles from LDS (stride 17 => conflict-free) and runs 4 WMMA chains.
// ---------------------------------------------------------------------------
__global__ void __launch_bounds__(512)
gemm_cplx_fused(const float* __restrict__ wr, const float* __restrict__ wi,
                const float* __restrict__ Ar, const float* __restrict__ Ai,
                const float* __restrict__ Br, const float* __restrict__ Bi,
                float* __restrict__ Yr, float* __restrict__ Yi,
                int ci, int co, int H2, int W2f, int m2)
{
    __shared__ float lw[2 * 8 * 16 * 17];   // [plane][kk][nn][yy pad 17]
    const int x    = blockIdx.x;
    const int y0   = blockIdx.y << 4;
    const int yy   = threadIdx.y;           // wave id = y within tile
    const int y    = y0 + yy;
    const int lane = threadIdx.x;
    const int half = lane >> 4;
    const int l    = lane & 15;
    const int t    = (threadIdx.y << 5) | lane;
    const bool valid = (y < m2);            // wave-uniform
    const size_t HW2 = (size_t)H2 * W2f;
    const int loc  = x * m2 + (valid ? y : 0);
    const int ntiles = co >> 4;

    const float* __restrict__ Abr = Ar + (size_t)loc * ci * 16;
    const float* __restrict__ Abi = Ai + (size_t)loc * ci * 16;

    for (int nt = 0; nt < ntiles; ++nt) {
        const int n0 = nt << 4;
        v8f acc_rr = {}, acc_ii = {}, acc_ri = {}, acc_ir = {};

        for (int k0 = 0; k0 < ci; k0 += 8) {
            const int kcnt = (ci - k0 < 8) ? (ci - k0) : 8;   // 8 or 4
            __syncthreads();   // previous stage fully consumed
#pragma unroll
            for (int j = 0; j < 8; ++j) {
                const int dwi   = t + (j << 9);     // 0..4095
                const int ly    = dwi & 15;
                const int r     = dwi >> 4;         // 0..255
                const int plane = r >> 7;
                const int rc    = r & 127;
                const int kk    = rc >> 4, nn = rc & 15;
                if (kk < kcnt && (y0 + ly) < m2) {
                    const float* src = (plane ? wi : wr)
                        + (size_t)((k0 + kk) * co + n0 + nn) * HW2
                        + (size_t)x * W2f + y0 + ly;
                    float* dst = &lw[(((plane << 3) + kk) * 16 + nn) * 17 + ly];
#if HAVE_ASYNC_LDS
                    __builtin_amdgcn_global_load_async_to_lds_b32(
                        (as1_int*)(size_t)src,
                        (as3_int*)(unsigned int)(size_t)dst,
                        0, 0);
#else
                    *dst = *src;
#endif
                }
            }
#if HAVE_ASYNC_LDS
#if defined(__has_builtin)
#if __has_builtin(__builtin_amdgcn_s_wait_asynccnt)
            __builtin_amdgcn_s_wait_asynccnt(0);
#else
            asm volatile("s_wait_asynccnt 0x0" ::: "memory");
#endif
#else
            asm volatile("s_wait_asynccnt 0x0" ::: "memory");
#endif
#endif
            __syncthreads();   // stage visible to all waves

            if (valid) {
                for (int kh = 0; kh < kcnt; kh += 4) {
                    const int ka = kh + (half << 1);
                    v2f a_r = { Abr[(k0 + ka) * 16 + l], Abr[(k0 + ka + 1) * 16 + l] };
                    v2f a_i = { Abi[(k0 + ka) * 16 + l], Abi[(k0 + ka + 1) * 16 + l] };
                    v2f b_r = { lw[(ka * 16 + l) * 17 + yy],
                                lw[((ka + 1) * 16 + l) * 17 + yy] };
                    v2f b_i = { lw[(((8 + ka) * 16) + l) * 17 + yy],
                                lw[(((8 + ka + 1) * 16) + l) * 17 + yy] };
                    acc_rr = WMMA_F32(a_r, b_r, acc_rr);
                    acc_ii = WMMA_F32(a_i, b_i, acc_ii);
                    acc_ri = WMMA_F32(a_r, b_i, acc_ri);
                    acc_ir = WMMA_F32(a_i, b_r, acc_ir);
                }
            }
        }

        if (valid) {
            const int oc = n0 + l;
            const size_t bidx = ((size_t)oc * H2 + x) * W2f + y;
            const float bre = Br[bidx], bim = Bi[bidx];
            float* __restrict__ Yrp = Yr + ((size_t)loc * co + n0) * 16;
            float* __restrict__ Yip = Yi + ((size_t)loc * co + n0) * 16;
#pragma unroll
            for (int v = 0; v < 8; ++v) {
                const int m = v + (half << 3);
                Yrp[l * 16 + m] = acc_rr[v] - acc_ii[v] + bre;
                Yip[l * 16 + m] = acc_ri[v] + acc_ir[v] + bim;
            }
        }
    }
}

// ---------------------------------------------------------------------------
// Pack l0 input: concat(x + 1.0 (real only), K) -> [loc][76][16], loc over 128x65.
// ---------------------------------------------------------------------------
__global__ void pack_x0(const float* __restrict__ xr, const float* __restrict__ xi,
                        const float* __restrict__ Kr, const float* __restrict__ Ki,
                        float* __restrict__ Xr, float* __restrict__ Xi)
{
    __shared__ float tr[16][17];
    __shared__ float ti[16][17];
    const int c = blockIdx.y;            // 0..75
    const int loc0 = blockIdx.x << 4;
    const int t = threadIdx.x;
    const int bb = t >> 4, l = t & 15;
    const int HW = 8320;

    float vr, vi;
    if (c < 64) {
        const size_t s = (size_t)(bb * 64 + c) * HW + loc0 + l;
        vr = xr[s] + 1.0f;   // x + 1.0 adds to real part only
        vi = xi[s];
    } else {
        const size_t s = (size_t)(bb * 12 + (c - 64)) * HW + loc0 + l;
        vr = Kr[s];
        vi = Ki[s];
    }
    tr[bb][l] = vr; ti[bb][l] = vi;
    __syncthreads();
    const int l2 = t >> 4, b2 = t & 15;
    const size_t d = ((size_t)(loc0 + l2) * 76 + c) * 16 + b2;
    Xr[d] = tr[b2][l2]; Xi[d] = ti[b2][l2];
}

// ---------------------------------------------------------------------------
// Gather down-layer input (top m1 / bottom m1 rows, first m2 cols) + K channels.
// ---------------------------------------------------------------------------
__global__ void gather_down(const float* __restrict__ Ypr, const float* __restrict__ Ypi,
                            const float* __restrict__ X0r, const float* __restrict__ X0i,
                            float* __restrict__ Xr, float* __restrict__ Xi,
                            int m1, int m2, int Cprev, int Hprev, int Wprev,
                            int ci, long total)
{
    const long idx = (long)blockIdx.x * blockDim.x + threadIdx.x;
    if (idx >= total) return;
    const int b = (int)(idx & 15);
    const long t = idx >> 4;
    const int c = (int)(t % ci);
    const int loc2 = (int)(t / ci);
    const int yy = loc2 % m2, xx = loc2 / m2;
    float r, im;
    if (c < Cprev) {
        const int xs = (xx < m1) ? xx : (Hprev - 2 * m1 + xx);
        const size_t s = ((size_t)(xs * Wprev + yy) * Cprev + c) * 16 + b;
        r = Ypr[s]; im = Ypi[s];
    } else {
        const int xs = (xx < m1) ? xx : (128 - 2 * m1 + xx);
        const size_t s = ((size_t)(xs * 65 + yy) * 76 + (64 + c - Cprev)) * 16 + b;
        r = X0r[s]; im = X0i[s];
    }
    Xr[idx] = r; Xi[idx] = im;
}

// ---------------------------------------------------------------------------
// Up-path scatter add.
// ---------------------------------------------------------------------------
__global__ void scatter_add(const float* __restrict__ Ur, const float* __restrict__ Ui,
                            float* __restrict__ Yr, float* __restrict__ Yi,
                            int m1, int m2, int Cprev, int Hprev, int Wprev, long total)
{
    const long idx = (long)blockIdx.x * blockDim.x + threadIdx.x;
    if (idx >= total) return;
    const int b = (int)(idx & 15);
    const long t = idx >> 4;
    const int c = (int)(t % Cprev);
    const int loc2 = (int)(t / Cprev);
    const int yy = loc2 % m2, xx = loc2 / m2;
    const int xs = (xx < m1) ? xx : (Hprev - 2 * m1 + xx);
    const size_t d = ((size_t)(xs * Wprev + yy) * Cprev + c) * 16 + b;
    Yr[d] += Ur[idx];
    Yi[d] += Ui[idx];
}

// ---------------------------------------------------------------------------
// GroupNorm (per batch over C*H*W): deterministic two-pass reduction.
// ---------------------------------------------------------------------------
__global__ void gn_partial(const float* __restrict__ Yr, const float* __restrict__ Yi,
                           float* __restrict__ part, long n16)
{
    const int t = threadIdx.x;
    const int b = t & 15, g = t >> 4;
    float sr = 0.f, srr = 0.f, si = 0.f, sii = 0.f;
    for (long r = (long)blockIdx.x * 16 + g; r < n16; r += (long)gridDim.x * 16) {
        const float vr = Yr[r * 16 + b];
        const float vi = Yi[r * 16 + b];
        sr += vr; srr += vr * vr; si += vi; sii += vi * vi;
    }
    __shared__ float red[16][16][4];
    red[g][b][0] = sr; red[g][b][1] = srr; red[g][b][2] = si; red[g][b][3] = sii;
    __syncthreads();
    if (t < 64) {
        const int b2 = t >> 2, st = t & 3;
        float s = 0.f;
        for (int gg = 0; gg < 16; ++gg) s += red[gg][b2][st];
        part[((size_t)blockIdx.x * 16 + b2) * 4 + st] = s;
    }
}

__global__ void gn_final(const float* __restrict__ part, float* __restrict__ stats,
                         int nblk, float invN)
{
    const int t = threadIdx.x;
    __shared__ float acc[16][4];
    if (t < 64) {
        const int b = t >> 2, st = t & 3;
        float s = 0.f;
        for (int k = 0; k < nblk; ++k) s += part[((size_t)k * 16 + b) * 4 + st];
        acc[b][st] = s;
    }
    __syncthreads();
    if (t < 16) {
        const float mur = acc[t][0] * invN;
        const float varr = acc[t][1] * invN - mur * mur;
        const float mui = acc[t][2] * invN;
        const float vari = acc[t][3] * invN - mui * mui;
        float* o = stats + t * 4;
        o[0] = mur; o[1] = rsqrtf(varr + 1e-5f);
        o[2] = mui; o[3] = rsqrtf(vari + 1e-5f);
    }
}

__global__ void gn_apply(float* __restrict__ Yr, float* __restrict__ Yi,
                         const float* __restrict__ stats,
                         const float* __restrict__ gr, const float* __restrict__ gbr,
                         const float* __restrict__ gi, const float* __restrict__ gbi,
                         int C, long total, float add1)
{
    const long idx = (long)blockIdx.x * blockDim.x + threadIdx.x;
    if (idx >= total) return;
    const int b = (int)(idx & 15);
    const int c = (int)((idx >> 4) % C);
    const float* st = stats + b * 4;
    const float vr = (Yr[idx] - st[0]) * st[1] * gr[c] + gbr[c];
    const float vi = (Yi[idx] - st[2]) * st[3] * gi[c] + gbi[c];
    Yr[idx] = gelu_exact(vr) + add1;   // "+1.0" applies to real part only
    Yi[idx] = gelu_exact(vi);
}

// ---------------------------------------------------------------------------
// Unpack [loc][C][16] planar r/i -> (b, c, x, y) complex64 (r,i interleaved).
// ---------------------------------------------------------------------------
__global__ void unpack_out(const float* __restrict__ Yr, const float* __restrict__ Yi,
                           float* __restrict__ out, int C, int S)
{
    __shared__ float tr[16][17];
    __shared__ float ti[16][17];
    const int c = blockIdx.y;
    const int loc0 = blockIdx.x << 4;
    const int t = threadIdx.x;
    const int l = t >> 4, bb = t & 15;
    const size_t s = ((size_t)(loc0 + l) * C + c) * 16 + bb;
    tr[l][bb] = Yr[s]; ti[l][bb] = Yi[s];
    __syncthreads();
    const int l2 = t & 15, b2 = t >> 4;
    const size_t d = (((size_t)(b2 * C + c)) * S + loc0 + l2) * 2;
    out[d] = tr[l2][b2];
    out[d + 1] = ti[l2][b2];
}

// ---------------------------------------------------------------------------
// Host orchestration
// ---------------------------------------------------------------------------
extern "C" void kernel_launch(void* const* d_in, const int* in_sizes, int n_in,
                              void* d_out, int out_size, void* d_ws, size_t ws_size,
                              hipStream_t stream)
{
    (void)in_sizes; (void)n_in; (void)out_size; (void)ws_size;

    const float* x_r = (const float*)d_in[0];
    const float* x_i = (const float*)d_in[1];
    const float* K_r = (const float*)d_in[2];
    const float* K_i = (const float*)d_in[3];

    // params flattened alphabetically: down1,down2,down3,l0,up1,up2,up3;
    // within each layer: b_i,b_r,gni_b,gni_w,gnr_b,gnr_w,w_i,w_r
    enum { B_I = 0, B_R = 1, GNI_B = 2, GNI_W = 3, GNR_B = 4, GNR_W = 5, W_I = 6, W_R = 7 };
    const int L_D1 = 4, L_D2 = 12, L_D3 = 20, L_L0 = 28, L_U1 = 36, L_U2 = 44, L_U3 = 52;
    auto P = [&](int base, int off) { return (const float*)d_in[base + off]; };

    const int S0 = 8320, S1 = 2112, S2 = 544, S3 = 144;

    float* ws = (float*)d_ws;
    size_t off = 0;
    auto alloc = [&](size_t n) { float* p = ws + off; off += n; return p; };
    float *X0r = alloc(10117120), *X0i = alloc(10117120);   // [8320][76][16]
    float *Y0r = alloc(8519680),  *Y0i = alloc(8519680);    // s0 [8320][64][16]
    float *X1r = alloc(2568192),  *X1i = alloc(2568192);    // [2112][76][16]
    float *Y1r = alloc(3244032),  *Y1i = alloc(3244032);    // s1 [2112][96][16]
    float *X2r = alloc(940032),   *X2i = alloc(940032);     // [544][108][16]
    float *Y2r = alloc(1114112),  *Y2i = alloc(1114112);    // s2 [544][128][16]
    float *X3r = alloc(322560),   *X3i = alloc(322560);     // [144][140][16]
    float *Y3r = alloc(368640),   *Y3i = alloc(368640);     // s3 [144][160][16]
    float *U3r = alloc(294912),   *U3i = alloc(294912);     // up3 out
    float *U2r = alloc(835584),   *U2i = alloc(835584);     // up2 out
    float *U1r = alloc(2162688),  *U1i = alloc(2162688);    // up1 out
    float *part = alloc(4096), *stats = alloc(64);

    auto runFused = [&](int base, const float* Ar, const float* Ai,
                        float* Yr, float* Yi, int ci, int co,
                        int H2, int W2f, int m2) {
        dim3 grid(H2, (m2 + 15) / 16);
        dim3 blk(32, 16);
        gemm_cplx_fused<<<grid, blk, 0, stream>>>(
            P(base, W_R), P(base, W_I), Ar, Ai, P(base, B_R), P(base, B_I),
            Yr, Yi, ci, co, H2, W2f, m2);
    };
    auto runGN = [&](float* Yr, float* Yi, int base, int C, int S, float add1) {
        const long n16 = (long)S * C;
        gn_partial<<<64, 256, 0, stream>>>(Yr, Yi, part, n16);
        gn_final<<<1, 64, 0, stream>>>(part, stats, 64, 1.0f / (float)n16);
        const long total = n16 * 16;
        gn_apply<<<(total + 255) / 256, 256, 0, stream>>>(
            Yr, Yi, stats, P(base, GNR_W), P(base, GNR_B), P(base, GNI_W), P(base, GNI_B),
            C, total, add1);
    };

    // ---- l0 ----
    {
        dim3 g(S0 / 16, 76);
        pack_x0<<<g, 256, 0, stream>>>(x_r, x_i, K_r, K_i, X0r, X0i);
    }
    {
        dim3 blk(32, 4);   // co=64 -> 4 n-tiles
        gemm_cplx_shared<<<S0, blk, 0, stream>>>(
            X0r, X0i, P(L_L0, W_R), P(L_L0, W_I), P(L_L0, B_R), P(L_L0, B_I),
            Y0r, Y0i, 76, 64);
    }
    runGN(Y0r, Y0i, L_L0, 64, S0, 0.0f);

    // ---- down chain ----
    struct Lvl { int m1, m2, W2f, Cprev, Hprev, Wprev, ci, co, S; };
    const Lvl D[3] = {
        {32, 33, 34,  64, 128, 65,  76,  96, S1},   // down1
        {16, 17, 18,  96,  64, 33, 108, 128, S2},   // down2
        { 8,  9, 10, 128,  32, 17, 140, 160, S3},   // down3
    };
    const int DBASE[3] = {L_D1, L_D2, L_D3};
    float* Xr[3] = {X1r, X2r, X3r};  float* Xi_[3] = {X1i, X2i, X3i};
    float* Yr_[4] = {Y0r, Y1r, Y2r, Y3r};  float* Yi_[4] = {Y0i, Y1i, Y2i, Y3i};

    for (int i = 0; i < 3; ++i) {
        const Lvl& L = D[i];
        const long gt = (long)L.S * L.ci * 16;
        gather_down<<<(gt + 255) / 256, 256, 0, stream>>>(
            Yr_[i], Yi_[i], X0r, X0i, Xr[i], Xi_[i],
            L.m1, L.m2, L.Cprev, L.Hprev, L.Wprev, L.ci, gt);
        runFused(DBASE[i], Xr[i], Xi_[i], Yr_[i + 1], Yi_[i + 1],
                 L.ci, L.co, 2 * L.m1, L.W2f, L.m2);
        runGN(Yr_[i + 1], Yi_[i + 1], DBASE[i], L.co, L.S, 1.0f);   // +1.0 after down
    }

    // ---- up chain: i = 3,2,1 ----
    struct Up { int m1, m2, W2f, ci, co, S, Cprev, Hprev, Wprev; };
    const Up U[3] = {
        { 8,  9, 10, 160, 128, S3, 128, 32, 17},    // up3: s3 -> add into s2
        {16, 17, 18, 128,  96, S2,  96, 64, 33},    // up2
        {32, 33, 34,  96,  64, S1,  64, 128, 65},   // up1
    };
    const int UBASE[3] = {L_U3, L_U2, L_U1};
    float* Ur_[3] = {U3r, U2r, U1r};  float* Ui_[3] = {U3i, U2i, U1i};

    for (int j = 0; j < 3; ++j) {
        const Up& L = U[j];
        const int lvl = 3 - j;
        runFused(UBASE[j], Yr_[lvl], Yi_[lvl], Ur_[j], Ui_[j],
                 L.ci, L.co, 2 * L.m1, L.W2f, L.m2);
        runGN(Ur_[j], Ui_[j], UBASE[j], L.co, L.S, 0.0f);
        const long at = (long)L.S * L.co * 16;
        scatter_add<<<(at + 255) / 256, 256, 0, stream>>>(
            Ur_[j], Ui_[j], Yr_[lvl - 1], Yi_[lvl - 1],
            L.m1, L.m2, L.Cprev, L.Hprev, L.Wprev, at);
    }

    // ---- outputs: (s0, s0, s1, s2, s3) as complex64 (r,i interleaved) ----
    float* out = (float*)d_out;
    size_t o = 0;
    auto emit = [&](const float* yr, const float* yi, int C, int S) {
        dim3 g(S / 16, C);
        unpack_out<<<g, 256, 0, stream>>>(yr, yi, out + o, C, S);
        o += (size_t)16 * C * S * 2;
    };
    emit(Y0r, Y0i, 64, S0);
    emit(Y0r, Y0i, 64, S0);
    emit(Y1r, Y1i, 96, S1);
    emit(Y2r, Y2i, 128, S2);
    emit(Y3r, Y3i, 160, S3);
}